// SVF_2D_52922587022056
// MI455X (gfx1250) — compile-verified
//
#include <hip/hip_runtime.h>
#include <hip/hip_bf16.h>
#include <stdint.h>

// SVF scaling-and-squaring, N=32 C=2 H=512 W=512, 16 steps.
// Memory-bound gather kernel; ping-pong buffers (2x67MB) fit in MI455X 192MB L2.
// CDNA5 paths: global_load_async_to_lds_b32/b64 (ASYNCcnt) for streaming
// operands, TH_STORE_WB scope:SCOPE_DEV stores to pin intermediates in L2
// (default RT stores wr-rinse to HBM), wave32 tiling (256 thr = 8 waves).

#define NB   32
#define HH   512
#define WW   512
#define HW   (HH * WW)          // 262144 = 2^18
#define NHW  (NB * HW)          // 8388608
#define NCHW (NB * 2 * HW)      // 16777216 floats per output tensor

typedef float v4f __attribute__((ext_vector_type(4)));   // lowers to a VGPR quad

// ---------------- CDNA5 async global->LDS + cache-hint helpers ----------------

__device__ __forceinline__ unsigned lds_off(const void* p) {
    // AMDGPU generic pointers to LDS carry the LDS byte offset in the low 32 bits.
    return (unsigned)(uintptr_t)p;
}

__device__ __forceinline__ void async_load_b32(const float* g, unsigned loff) {
    asm volatile("global_load_async_to_lds_b32 %0, %1, off"
                 :: "v"(loff), "v"(g) : "memory");
}

__device__ __forceinline__ void async_load_b64(const float* g, unsigned loff) {
    asm volatile("global_load_async_to_lds_b64 %0, %1, off"
                 :: "v"(loff), "v"(g) : "memory");
}

__device__ __forceinline__ void wait_async0() {
    asm volatile("s_wait_asynccnt 0" ::: "memory");
}

// Store that completes at device-coherent L2 and stays dirty there
// (TH=WB overrides the default RT wr-rinse => no background HBM writeback
// for the 15 intermediate warp-field generations).
__device__ __forceinline__ void store_b32_wb(float* g, float d) {
    asm volatile("global_store_b32 %0, %1, off th:TH_STORE_WB scope:SCOPE_DEV"
                 :: "v"(g), "v"(d) : "memory");
}

__device__ __forceinline__ void store_b128_wb(v4f* g, v4f d) {
    asm volatile("global_store_b128 %0, %1, off th:TH_STORE_WB scope:SCOPE_DEV"
                 :: "v"(g), "v"(d) : "memory");
}

// ---------------- kernels ----------------

// wf0 = v * 2^-16  (vectorized streaming; result pinned in L2)
__global__ __launch_bounds__(256) void svf_init(const v4f* __restrict__ v,
                                                v4f* __restrict__ wf) {
    unsigned i = blockIdx.x * 256u + threadIdx.x;   // < NCHW/4
    v4f a = v[i] * (1.0f / 65536.0f);
    store_b128_wb(&wf[i], a);
}

// One scaling-and-squaring step: wf_out = wf_in + bilinear(wf_in, id + wf_in)
__global__ __launch_bounds__(256) void svf_step(const float* __restrict__ wfin,
                                                float* __restrict__ wfout,
                                                const float* __restrict__ idg) {
    __shared__ float s_wfx[256];
    __shared__ float s_wfy[256];
    __shared__ float s_id[512];   // interleaved (gx, gy) identity pairs

    const int lane = threadIdx.x;
    const unsigned t = blockIdx.x * 256u + lane;    // < NHW
    const int n = (int)(t >> 18);                   // t / HW
    const int p = (int)(t & (HW - 1));              // t % HW (same row per block)

    const float* __restrict__ base = wfin + ((size_t)n << 19);   // n*2*HW
    const float* __restrict__ c0   = base;                       // x-displacement plane
    const float* __restrict__ c1   = base + HW;                  // y-displacement plane

    // Stage streaming operands into LDS asynchronously (overlaps address math).
    async_load_b32(c0 + p,       lds_off(&s_wfx[lane]));
    async_load_b32(c1 + p,       lds_off(&s_wfy[lane]));
    async_load_b64(idg + 2 * p,  lds_off(&s_id[2 * lane]));
    wait_async0();
    // Each lane consumes only data issued by its own wave: no barrier required.

    const float wfx = s_wfx[lane];
    const float wfy = s_wfy[lane];
    const float gx  = s_id[2 * lane + 0] + wfx;
    const float gy  = s_id[2 * lane + 1] + wfy;

    // torch grid_sample: align_corners=False, padding_mode='border'
    float ix = 0.5f * ((gx + 1.0f) * (float)WW - 1.0f);
    float iy = 0.5f * ((gy + 1.0f) * (float)HH - 1.0f);
    ix = fminf(fmaxf(ix, 0.0f), (float)(WW - 1));
    iy = fminf(fmaxf(iy, 0.0f), (float)(HH - 1));

    const float fx0 = floorf(ix), fy0 = floorf(iy);
    const float wx = ix - fx0,    wy = iy - fy0;
    const int x0 = (int)fx0,      y0 = (int)fy0;
    const int x1 = min(x0 + 1, WW - 1);
    const int y1 = min(y0 + 1, HH - 1);

    const int i00 = y0 * WW + x0;
    const int i01 = y0 * WW + x1;
    const int i10 = y1 * WW + x0;
    const int i11 = y1 * WW + x1;

    // Gathers: all hit the 192MB device L2 (working set is L2-resident).
    float a00 = c0[i00], a01 = c0[i01], a10 = c0[i10], a11 = c0[i11];
    float b00 = c1[i00], b01 = c1[i01], b10 = c1[i10], b11 = c1[i11];

    const float atop = a00 + (a01 - a00) * wx;
    const float abot = a10 + (a11 - a10) * wx;
    const float sa   = atop + (abot - atop) * wy;

    const float btop = b00 + (b01 - b00) * wx;
    const float bbot = b10 + (b11 - b10) * wx;
    const float sb   = btop + (bbot - btop) * wy;

    float* __restrict__ obase = wfout + ((size_t)n << 19);
    store_b32_wb(obase + p,      wfx + sa);
    store_b32_wb(obase + HW + p, wfy + sb);
}

// transformation = wf + transpose(identity, NHWC->NCHW); true output -> NT store
__global__ __launch_bounds__(256) void svf_final(const float* __restrict__ wf,
                                                 float* __restrict__ trans,
                                                 const float* __restrict__ idg) {
    const unsigned t = blockIdx.x * 256u + threadIdx.x;   // < NCHW
    const int p = (int)(t & (HW - 1));
    const int c = (int)((t >> 18) & 1);
    const float idv = idg[2 * p + c];
    __builtin_nontemporal_store(wf[t] + idv, &trans[t]);
}

// ---------------- launch ----------------

extern "C" void kernel_launch(void* const* d_in, const int* in_sizes, int n_in,
                              void* d_out, int out_size, void* d_ws, size_t ws_size,
                              hipStream_t stream) {
    const float* v   = (const float*)d_in[0];        // [32, 2, 512, 512]
    const float* idg = (const float*)d_in[1];        // [1, 512, 512, 2]
    float* P = (float*)d_out;                        // transformation slot
    float* Q = P + (size_t)NCHW;                     // warp_field slot

    // Ping-pong inside d_out: init -> Q; 16 steps (even) end back in Q.
    svf_init<<<NCHW / 4 / 256, 256, 0, stream>>>((const v4f*)v, (v4f*)Q);

    float* a = Q;
    float* b = P;
    for (int s = 0; s < 16; ++s) {
        svf_step<<<NHW / 256, 256, 0, stream>>>(a, b, idg);
        float* tmp = a; a = b; b = tmp;
    }
    // a == Q holds the final warp_field (correct output slot).

    svf_final<<<NCHW / 256, 256, 0, stream>>>(Q, P, idg);
}